// SinkhornEval_14654428414010
// MI455X (gfx1250) — compile-verified
//
#include <hip/hip_runtime.h>
#include <math.h>

// Problem constants from the reference setup_inputs(): N=4, P=Q=4096, D=4.
#define NB 4
#define PP 4096
#define QQ 4096
#define EPSV 0.01f
#define INV_EPS 100.0f
#define CLIPV (1.0f - 1e-6f)

typedef __attribute__((ext_vector_type(2))) float v2f;
typedef __attribute__((ext_vector_type(8))) float v8f;

// ---------------------------------------------------------------------------
// Cost construction: one wave computes a 16x16 tile of quaternion dots with
// V_WMMA_F32_16X16X4_F32 (K=4 == quaternion dim), then applies
// -C/eps = -(2/eps)*acos(min(|dot|, CLIP)) and stores it (plus optionally the
// transpose, for the column-reduction of the asymmetric xy problem).
//
// A operand (16x4 f32): lane L holds row M=L&15, K = 2*(L>>4) + {0,1} -> v2f.
// B operand (4x16 f32): lane L holds col N=L&15, same K striping    -> v2f.
// D (16x16 f32): vgpr r, lane L -> element (M = r + 8*(L>>4), N = L&15).
// ---------------------------------------------------------------------------
__global__ __launch_bounds__(256) void cost_tiles(const float* __restrict__ X,
                                                  const float* __restrict__ Y,
                                                  float* __restrict__ C,
                                                  float* __restrict__ CT) {
    const int lane = threadIdx.x & 31;
    const int wv   = threadIdx.x >> 5;
    int tile = blockIdx.x * 8 + wv;             // N*(P/16)*(Q/16) tiles total
    const int tj = tile & 255; tile >>= 8;
    const int ti = tile & 255;
    const int n  = tile >> 8;
    const int l15 = lane & 15;
    const int hi  = lane >> 4;

    const float* Xb = X + (size_t)n * PP * 4;
    const float* Yb = Y + (size_t)n * QQ * 4;
    v2f a = *(const v2f*)(Xb + (size_t)(ti * 16 + l15) * 4 + hi * 2);
    v2f b = *(const v2f*)(Yb + (size_t)(tj * 16 + l15) * 4 + hi * 2);

    v8f acc = {};
    // 8 args: (neg_a, A, neg_b, B, c_mod, C, reuse_a, reuse_b)
    acc = __builtin_amdgcn_wmma_f32_16x16x4_f32(false, a, false, b,
                                                (short)0, acc, false, false);

    float vals[8];
#pragma unroll
    for (int r = 0; r < 8; ++r) {
        float t = fminf(fabsf(acc[r]), CLIPV);
        vals[r] = (-2.0f * INV_EPS) * acosf(t);     // store -C/eps
    }

    const int col  = tj * 16 + l15;
    const int row0 = ti * 16 + hi * 8;
#pragma unroll
    for (int r = 0; r < 8; ++r)
        C[((size_t)n * PP + row0 + r) * QQ + col] = vals[r];

    if (CT) {   // CT[n][col][row]: per lane, 8 consecutive rows -> 2x float4
        float* p = CT + ((size_t)n * QQ + col) * PP + row0;
        *(float4*)(p)     = make_float4(vals[0], vals[1], vals[2], vals[3]);
        *(float4*)(p + 4) = make_float4(vals[4], vals[5], vals[6], vals[7]);
    }
}

// ---------------------------------------------------------------------------
// One Sinkhorn half-iteration. One wave per row (4096 elements):
//   out_i  = -eps * LSE_j( t_in[j] + negCeps[i][j] )
//   t_out_i = log(w_out[i]) + out_i / eps        (feeds the next half-iter)
// Two passes (max, then sum-exp); the 16KB row stays resident in cache
// between passes. Coalesced float4 loads, wave32 shuffle reductions.
// ---------------------------------------------------------------------------
__global__ __launch_bounds__(256) void lse_rows(const float* __restrict__ C,
                                                const float* __restrict__ t_in,
                                                const float* __restrict__ w_out,
                                                float* __restrict__ out,
                                                float* __restrict__ t_out) {
    const int lane  = threadIdx.x & 31;
    const int wv    = threadIdx.x >> 5;
    const int rowid = blockIdx.x * 8 + wv;       // 0 .. NB*4096-1
    const int n     = rowid >> 12;
    const float* crow = C + (size_t)rowid * 4096;
    const float* trow = t_in + (size_t)n * 4096;

    float m = -3.4e38f;
#pragma unroll 4
    for (int it = 0; it < 32; ++it) {
        const int idx = it * 128 + lane * 4;
        float4 c  = *(const float4*)(crow + idx);
        float4 tv = *(const float4*)(trow + idx);
        m = fmaxf(m, fmaxf(fmaxf(c.x + tv.x, c.y + tv.y),
                           fmaxf(c.z + tv.z, c.w + tv.w)));
    }
#pragma unroll
    for (int off = 16; off > 0; off >>= 1)
        m = fmaxf(m, __shfl_xor(m, off, 32));

    float s = 0.0f;
#pragma unroll 4
    for (int it = 0; it < 32; ++it) {
        const int idx = it * 128 + lane * 4;
        float4 c  = *(const float4*)(crow + idx);
        float4 tv = *(const float4*)(trow + idx);
        s += __expf(c.x + tv.x - m) + __expf(c.y + tv.y - m) +
             __expf(c.z + tv.z - m) + __expf(c.w + tv.w - m);
    }
#pragma unroll
    for (int off = 16; off > 0; off >>= 1)
        s += __shfl_xor(s, off, 32);

    if (lane == 0) {
        const float v = -EPSV * (m + __logf(s));
        out[rowid]   = v;
        t_out[rowid] = __logf(w_out[rowid]) + v * INV_EPS;
    }
}

// t_j = log(w_j) + g0_j/eps with g0 = 0.
__global__ void init_t(const float* __restrict__ w, float* __restrict__ t,
                       int n_total) {
    const int i = blockIdx.x * blockDim.x + threadIdx.x;
    if (i < n_total) t[i] = __logf(w[i]);
}

// sum_n [ <a,f_xy>+<b,g_xy> - 0.5(<a,f_xx>+<a,g_xx>) - 0.5(<b,f_yy>+<b,g_yy>) ]
__global__ __launch_bounds__(256) void finalize(const float* __restrict__ wx,
                                                const float* __restrict__ wy,
                                                const float* __restrict__ f_xy,
                                                const float* __restrict__ g_xy,
                                                const float* __restrict__ f_xx,
                                                const float* __restrict__ g_xx,
                                                const float* __restrict__ f_yy,
                                                const float* __restrict__ g_yy,
                                                float* __restrict__ outp) {
    __shared__ float red[256];
    float acc = 0.0f;
    for (int i = threadIdx.x; i < NB * PP; i += 256) {
        const float a = wx[i], b = wy[i];
        acc += a * f_xy[i] + b * g_xy[i]
             - 0.5f * (a * f_xx[i] + a * g_xx[i])
             - 0.5f * (b * f_yy[i] + b * g_yy[i]);
    }
    red[threadIdx.x] = acc;
    __syncthreads();
    for (int s = 128; s > 0; s >>= 1) {
        if (threadIdx.x < s) red[threadIdx.x] += red[threadIdx.x + s];
        __syncthreads();
    }
    if (threadIdx.x == 0) outp[0] = red[0];
}

extern "C" void kernel_launch(void* const* d_in, const int* in_sizes, int n_in,
                              void* d_out, int out_size, void* d_ws,
                              size_t ws_size, hipStream_t stream) {
    (void)in_sizes; (void)n_in; (void)out_size; (void)ws_size;
    const float* x  = (const float*)d_in[0];
    const float* wx = (const float*)d_in[1];
    const float* y  = (const float*)d_in[2];
    const float* wy = (const float*)d_in[3];

    float* ws = (float*)d_ws;
    const size_t MSZ = (size_t)NB * PP * QQ;     // 67,108,864 floats each
    float* Cxy = ws;
    float* Cyx = ws + MSZ;                       // transpose of Cxy
    float* Cxx = ws + 2 * MSZ;                   // symmetric
    float* Cyy = ws + 3 * MSZ;                   // symmetric
    float* sm  = ws + 4 * MSZ;

    const int V = NB * PP;                       // 16384 (== NB*QQ)
    float *f[3], *tf[3], *g[3], *tg[3];
    for (int o = 0; o < 3; ++o) {
        f[o]  = sm + (size_t)o * 4 * V;
        tf[o] = f[o]  + V;
        g[o]  = tf[o] + V;
        tg[o] = g[o]  + V;
    }

    const int cost_blocks = (NB * (PP / 16) * (QQ / 16)) / 8;   // 32768
    cost_tiles<<<cost_blocks, 256, 0, stream>>>(x, y, Cxy, Cyx);
    cost_tiles<<<cost_blocks, 256, 0, stream>>>(x, x, Cxx, nullptr);
    cost_tiles<<<cost_blocks, 256, 0, stream>>>(y, y, Cyy, nullptr);

    const float* Cf[3] = {Cxy, Cxx, Cyy};        // f-update matrices (row-LSE)
    const float* Cg[3] = {Cyx, Cxx, Cyy};        // g-update matrices (row-LSE on transpose)
    const float* w1[3] = {wx, wx, wy};           // weights along i
    const float* w2[3] = {wy, wx, wy};           // weights along j

    const int lse_blocks = (NB * PP) / 8;        // 2048 blocks, 1 wave/row
    for (int o = 0; o < 3; ++o) {
        init_t<<<V / 256, 256, 0, stream>>>(w2[o], tg[o], V);
        for (int it = 0; it < 10; ++it) {
            lse_rows<<<lse_blocks, 256, 0, stream>>>(Cf[o], tg[o], w1[o], f[o], tf[o]);
            lse_rows<<<lse_blocks, 256, 0, stream>>>(Cg[o], tf[o], w2[o], g[o], tg[o]);
        }
    }
    finalize<<<1, 256, 0, stream>>>(wx, wy, f[0], g[0], f[1], g[1], f[2], g[2],
                                    (float*)d_out);
}